// IRadon_4707284156588
// MI455X (gfx1250) — compile-verified
//
#include <hip/hip_runtime.h>

// ---------------------------------------------------------------------------
// Inverse Radon (filtered backprojection), MI455X / gfx1250.
//  Stage 0: build Ram-Lak Toeplitz matrix H[368][368] (K-edge col zeroed) and
//           the cos/sin table in workspace.
//  Stage 1: ramp filter as GEMM  C(363x180) = H(363x363) * X(363x180) per
//           batch on the fp32 matrix pipe (V_WMMA_F32_16X16X4_F32).
//           Branchless inner loop: 1x b64 A-load + 2x b32 B-loads + 1 WMMA.
//           Result stored transposed+padded [b][192][368] with two b128
//           stores so the backprojection gathers contiguous s.
//  Stage 2: backprojection, 180-angle loop, trig in LDS, provably in-range
//           bilinear gathers (no guards needed for the 256x256 crop).
// ---------------------------------------------------------------------------

typedef __attribute__((ext_vector_type(2))) float v2f;
typedef __attribute__((ext_vector_type(8))) float v8f;

#define GRID_N 363            // ceil(sqrt(2)*256)
#define NTH    180            // angles
#define NTHP   192            // angle dim padded to multiple of 16
#define SPAD   368            // s dim padded to multiple of 16
#define MT     23             // 368/16 M-tiles
#define NT     12             // 192/16 N-tiles
#define NBATCH 4
#define CROP0  53             // pad_before
#define PI_D   3.14159265358979323846

#define FT_ELEMS (NBATCH * NTHP * SPAD)  // transposed filtered sinogram
#define H_ELEMS  (SPAD * SPAD)           // Toeplitz ramp matrix

// ---- Stage 0a: Toeplitz Ram-Lak matrix -----------------------------------
__global__ void irk_init_h(float* __restrict__ H) {
    int idx = blockIdx.x * 256 + threadIdx.x;
    if (idx >= H_ELEMS) return;
    int m = idx / SPAD;
    int k = idx - m * SPAD;
    int d = m - k;
    // h(0)=0.5 ; odd d: -2/(pi^2 d^2) ; even d!=0: 0 ; col >= 363: 0
    float fd = (float)d;
    float odd = -0.2026423672846756f / (fd * fd);
    float v = (d == 0) ? 0.5f : ((d & 1) ? odd : 0.0f);
    H[idx] = (k < GRID_N) ? v : 0.0f;
}

// ---- Stage 0b: trig table -------------------------------------------------
__global__ void irk_trig(float* __restrict__ trig) {
    int a = threadIdx.x;
    if (a < NTH) {
        float th = (float)a * (float)(PI_D / 180.0);
        trig[2 * a]     = cosf(th);
        trig[2 * a + 1] = sinf(th);
    }
}

// ---- Stage 1: filter as WMMA GEMM ----------------------------------------
// one wave per 16x16 output tile; 4*23*12 = 1104 waves, 4 waves/block.
__global__ void __launch_bounds__(128) irk_filter_wmma(
        const float* __restrict__ x,   // [4][363][180]
        const float* __restrict__ H,   // [368][368]
        float* __restrict__ ft) {      // [4][192][368] transposed, padded
    const int lane = threadIdx.x & 31;
    const int wave = threadIdx.x >> 5;
    const int tile = blockIdx.x * 4 + wave;        // 0..1103

    const int b   = tile / (MT * NT);
    const int rem = tile % (MT * NT);
    const int mT  = rem / NT;
    const int nT  = rem % NT;

    const int l15  = lane & 15;
    const int kSel = (lane >> 4) << 1;             // 0 or 2
    const int row  = mT * 16 + l15;                // M (output s), 0..367
    const int col  = nT * 16 + l15;                // N (angle),    0..191
    // cols >= 180 only feed D columns that are never read: clamp, don't guard
    const int colL = (col < NTH) ? col : (NTH - 1);

    const float* __restrict__ ap = H + row * SPAD + kSel;
    const float* __restrict__ bp = x + b * (GRID_N * NTH) + kSel * NTH + colL;

    v8f acc = {0.f, 0.f, 0.f, 0.f, 0.f, 0.f, 0.f, 0.f};

    // main loop: kk+1 <= 359 < 363, fully unguarded. 90 WMMA steps.
    for (int k = 0; k < 360; k += 4) {
        v2f a = *(const v2f*)ap;                   // H[row][kk], H[row][kk+1]
        v2f bv;
        bv.x = bp[0];                              // X[kk][colL]
        bv.y = bp[NTH];                            // X[kk+1][colL]
        acc = __builtin_amdgcn_wmma_f32_16x16x4_f32(
                  false, a, false, bv, (short)0, acc, false, false);
        ap += 4;
        bp += 4 * NTH;
    }
    // tail k=360: kk = 360+kSel; for kSel==2, kk+1 = 363 is out of range ->
    // load row 362 again (offset 0); its coefficient H[row][363] is 0.
    {
        v2f a = *(const v2f*)ap;
        v2f bv;
        bv.x = bp[0];
        bv.y = bp[(kSel == 0) ? NTH : 0];
        acc = __builtin_amdgcn_wmma_f32_16x16x4_f32(
                  false, a, false, bv, (short)0, acc, false, false);
    }

    // store transposed: ft[b][col][M]; D layout: VGPR v -> M = mT*16 + v
    // (+8 for lanes 16..31), N = lane&15. Unconditional (angle dim padded).
    float* __restrict__ Fb = ft + b * (NTHP * SPAD) + col * SPAD + mT * 16
                             + ((lane >> 4) << 3);
    float4 lo = make_float4(acc[0], acc[1], acc[2], acc[3]);
    float4 hi = make_float4(acc[4], acc[5], acc[6], acc[7]);
    *(float4*)(Fb)     = lo;
    *(float4*)(Fb + 4) = hi;
}

// ---- Stage 2: backprojection over the 256x256 crop ------------------------
// For the crop, t in [-0.99621, 1.00011] -> pos in [0.68, 362.02],
// so i0 in [0,362] and i0+1 <= 363 < SPAD: reads stay inside the padded s row
// and the gathers need no guards.
__global__ void __launch_bounds__(256) irk_backproj(
        const float* __restrict__ ft,    // [4][192][368]
        const float* __restrict__ trig,  // [180][2]
        float* __restrict__ out) {       // [4][256][256]
    __shared__ float2 cs_lds[NTH];
    const int tid = threadIdx.x;
    if (tid < NTH) cs_lds[tid] = make_float2(trig[2 * tid], trig[2 * tid + 1]);
    __syncthreads();

    const int idx = blockIdx.x * 256 + tid;
    const int b = idx >> 16;
    const int p = (idx >> 8) & 255;
    const int q = idx & 255;

    // unit[i] = (i - 181)/181 on the 363-point grid; crop offset 53
    const float yg = (float)(p + CROP0 - 181) * (1.0f / 181.0f);
    const float xg = (float)(q + CROP0 - 181) * (1.0f / 181.0f);

    const float* __restrict__ Fa = ft + b * (NTHP * SPAD);
    float acc = 0.0f;
    #pragma unroll 4
    for (int a = 0; a < NTH; ++a) {
        const float2 cs  = cs_lds[a];
        const float  t   = xg * cs.x - yg * cs.y;
        const float  pos = fmaf(t, 181.0f, 181.0f);   // (t+1)*(S-1)/2
        const float  fi  = floorf(pos);
        const float  w   = pos - fi;
        const int    i0  = (int)fi;
        const float  v0  = Fa[i0];
        const float  v1  = Fa[i0 + 1];
        acc = fmaf(1.0f - w, v0, fmaf(w, v1, acc));
        Fa += SPAD;
    }
    out[idx] = acc * (float)(PI_D / (2.0 * NTH));
}

// ---------------------------------------------------------------------------
extern "C" void kernel_launch(void* const* d_in, const int* in_sizes, int n_in,
                              void* d_out, int out_size, void* d_ws, size_t ws_size,
                              hipStream_t stream) {
    const float* x  = (const float*)d_in[0];      // [4][1][363][180] fp32
    float* out      = (float*)d_out;              // [4][1][256][256] fp32
    float* ft       = (float*)d_ws;               // FT_ELEMS floats
    float* Hm       = ft + FT_ELEMS;              // H_ELEMS floats
    float* trig     = Hm + H_ELEMS;               // 360 floats
    (void)in_sizes; (void)n_in; (void)out_size; (void)ws_size;

    irk_init_h<<<(H_ELEMS + 255) / 256, 256, 0, stream>>>(Hm);
    irk_trig<<<1, 256, 0, stream>>>(trig);
    irk_filter_wmma<<<(NBATCH * MT * NT) / 4, 128, 0, stream>>>(x, Hm, ft);
    irk_backproj<<<(NBATCH * 256 * 256) / 256, 256, 0, stream>>>(ft, trig, out);
}